// BranchLinear_70677981823113
// MI455X (gfx1250) — compile-verified
//
#include <hip/hip_runtime.h>
#include <hip/hip_bf16.h>

typedef __attribute__((ext_vector_type(2))) float v2f;
typedef __attribute__((ext_vector_type(8))) float v8f;

#define TT 131072
#define DD 512
#define NB 8

#define TM 64     // rows per block tile
#define TNB 128   // cols per block tile
#define BK 64     // k-chunk
#define SA 68     // lds_a row stride (row-major [row][k]); banks 4m+k -> conflict-free b64 frags
#define SB 132    // lds_b row stride ([k][col]); 132*4 % 16 == 0 for b128 async stores

// ---------------- bucketing ----------------
__global__ void bl_zero(int* counts, int* cursor, float* zbuf) {
    if (threadIdx.x < NB) { counts[threadIdx.x] = 0; cursor[threadIdx.x] = 0; }
    if (threadIdx.x < BK) zbuf[threadIdx.x] = 0.f;
}

__global__ void bl_count(const int* __restrict__ idx, int* __restrict__ counts) {
    int t = blockIdx.x * blockDim.x + threadIdx.x;
    if (t < TT) atomicAdd(&counts[idx[t]], 1);
}

__global__ void bl_scan(const int* __restrict__ counts, int* __restrict__ offsets) {
    if (threadIdx.x == 0) {
        int s = 0;
        for (int i = 0; i < NB; ++i) { offsets[i] = s; s += counts[i]; }
    }
}

__global__ void bl_fill(const int* __restrict__ idx, const int* __restrict__ offsets,
                        int* __restrict__ cursor, int* __restrict__ perm) {
    int t = blockIdx.x * blockDim.x + threadIdx.x;
    if (t < TT) {
        int b = idx[t];
        int p = atomicAdd(&cursor[b], 1);
        perm[offsets[b] + p] = t;
    }
}

// ---- async global -> LDS copy of one 16B segment (CDNA5, tracked by ASYNCcnt) ----
__device__ __forceinline__ void async_b128(const float* gsrc, const float* ldsdst) {
    unsigned int loff = (unsigned int)(unsigned long long)ldsdst; // low 32b = LDS byte addr
    asm volatile("global_load_async_to_lds_b128 %0, %1, off"
                 :: "v"(loff), "v"(gsrc) : "memory");
}

__device__ __forceinline__ void wait_async0() {
    asm volatile("s_wait_asynccnt 0x0" ::: "memory");
}

// ---------------- per-branch tiled WMMA GEMM ----------------
// grid: (T/TM, D/TNB, NB), block: 256 threads = 8 waves
// wave w: row-slab rs = w>>1 (16 rows), col-slab cs = w&1 (64 cols) -> 4 WMMA tiles
__global__ void __launch_bounds__(256)
bl_gemm(const float* __restrict__ x, const float* __restrict__ w,
        const float* __restrict__ bias, const int* __restrict__ perm,
        const int* __restrict__ counts, const int* __restrict__ offsets,
        const float* __restrict__ zbuf, float* __restrict__ out)
{
    const int n    = blockIdx.z;
    const int cnt  = counts[n];                  // scalar per block -> uniform exit
    const int row0 = blockIdx.x * TM;
    if (row0 >= cnt) return;
    const int colb = blockIdx.y * TNB;
    const int offs = offsets[n];

    __shared__ float lds_a[2][TM * SA];          // [row][k], double-buffered
    __shared__ float lds_b[2][BK * SB];          // [k][col], double-buffered
    __shared__ int   row_g[TM];

    const int tid = threadIdx.x;
    if (tid < TM) {
        int r = row0 + tid;
        row_g[tid] = (r < cnt) ? perm[offs + r] : -1;
    }
    __syncthreads();

    const int wave  = tid >> 5;
    const int lane  = tid & 31;
    const int rs    = wave >> 1;                 // 0..3
    const int cs    = wave & 1;                  // 0..1
    const int mlo   = lane & 15;
    const int khalf = (lane >> 4) * 2;           // lanes 0-15 -> K0/K1, 16-31 -> K2/K3

    const float* wn = w + (size_t)n * DD * DD;

    // per-thread staging coordinates
    const int a_seg   = tid & 15;                // 16B segment within a row's k-chunk
    const int a_rbase = tid >> 4;                // 0..15
    const int b_seg   = tid & 31;                // 16B segment within a k-row of B
    const int b_kbase = tid >> 5;                // 0..7

    // issue one double-buffer stage: 4 A-segments + 8 B-segments per thread
    auto issue_stage = [&](int kk, int buf) {
        #pragma unroll
        for (int i = 0; i < 4; ++i) {
            int r = a_rbase + i * 16;
            int g = row_g[r];
            const float* src = (g >= 0) ? (x + (size_t)g * DD + kk) : zbuf;
            async_b128(src + a_seg * 4, &lds_a[buf][r * SA + a_seg * 4]);
        }
        #pragma unroll
        for (int i = 0; i < 8; ++i) {
            int k = b_kbase + i * 8;
            async_b128(wn + (size_t)(kk + k) * DD + colb + b_seg * 4,
                       &lds_b[buf][k * SB + b_seg * 4]);
        }
    };

    v8f acc[4] = {};

    issue_stage(0, 0);
    int cur = 0;
    for (int kk = 0; kk < DD; kk += BK) {
        wait_async0();
        __syncthreads();                         // stage `cur` fully resident
        if (kk + BK < DD) issue_stage(kk + BK, cur ^ 1);

        #pragma unroll 4
        for (int k4 = 0; k4 < BK; k4 += 4) {
            // A fragment: one b64 LDS load per lane, conflict-free (stride 68)
            v2f a = *(const v2f*)&lds_a[cur][(rs * 16 + mlo) * SA + k4 + khalf];
            #pragma unroll
            for (int t = 0; t < 4; ++t) {
                const int col = cs * 64 + t * 16 + mlo;
                v2f b;
                b.x = lds_b[cur][(k4 + khalf    ) * SB + col];
                b.y = lds_b[cur][(k4 + khalf + 1) * SB + col];
                acc[t] = __builtin_amdgcn_wmma_f32_16x16x4_f32(
                    false, a, false, b, (short)0, acc[t], false, false);
            }
        }
        __syncthreads();                         // all reads of `cur` done before overwrite
        cur ^= 1;
    }

    // ---- bias add + scattered store ----
    #pragma unroll
    for (int t = 0; t < 4; ++t) {
        const int col = colb + cs * 64 + t * 16 + mlo;
        const float bv = bias[n * DD + col];
        #pragma unroll
        for (int j = 0; j < 8; ++j) {
            int ml = rs * 16 + j + 8 * (lane >> 4);   // C layout: VGPR j, lane half
            int g = row_g[ml];
            if (g >= 0) out[(size_t)g * DD + col] = acc[t][j] + bv;
        }
    }
}

extern "C" void kernel_launch(void* const* d_in, const int* in_sizes, int n_in,
                              void* d_out, int out_size, void* d_ws, size_t ws_size,
                              hipStream_t stream) {
    const float* x    = (const float*)d_in[0];   // (T, D)
    const int*   idx  = (const int*)  d_in[1];   // (T,)
    const float* w    = (const float*)d_in[2];   // (N, D, D)
    const float* bias = (const float*)d_in[3];   // (N, D)
    float* out = (float*)d_out;

    int*   counts  = (int*)d_ws;          // NB
    int*   offsets = counts + NB;         // NB
    int*   cursor  = offsets + NB;        // NB
    int*   perm    = cursor + NB;         // TT
    float* zbuf    = (float*)(perm + TT); // BK zeros (async source for padded rows)

    bl_zero<<<1, 128, 0, stream>>>(counts, cursor, zbuf);
    bl_count<<<TT / 256, 256, 0, stream>>>(idx, counts);
    bl_scan<<<1, 32, 0, stream>>>(counts, offsets);
    bl_fill<<<TT / 256, 256, 0, stream>>>(idx, offsets, cursor, perm);

    dim3 grid(TT / TM, DD / TNB, NB);
    bl_gemm<<<grid, 256, 0, stream>>>(x, w, bias, perm, counts, offsets, zbuf, out);
}